// CA_28063316312676
// MI455X (gfx1250) — compile-verified
//
#include <hip/hip_runtime.h>

// MI455X / gfx1250, wave32. Grouped vector attention:
//   Q,K,V = x @ W^T (f32 WMMA 16x16x4 GEMM)
//   scores[b,i,j,h] = relu( sum_g relu(Q[b,i,h*16+g]-K[b,j,h*16+g]) * w[g] + bias )
//   att = softmax_j(scores)  (flash-style online, fused)
//   out[b,i,g*16+h] = sum_j att[b,i,j,h] * V[b,j,g*16+h]   (f32 WMMA 16x16x4)

typedef float v2f __attribute__((ext_vector_type(2)));
typedef float v8f __attribute__((ext_vector_type(8)));

#define NB      2
#define L_SEQ   512
#define D_MODEL 256
#define NHEAD   16
#define NGROUP  16

// ---------------------------------------------------------------------------
// Kernel 1: QKV projections. One wave -> one 16x16 tile of one of Q/K/V.
// y[r,e] = sum_d x[r,d] * W[e,d]
// A tile (16x4 of x): lane=M(rows), a[v] = x[row0+lane%16][d0 + v + 2*(lane/16)]
// B tile (4x16 of W^T): lane=N(cols=e), b[v] = W[e0+lane%16][d0 + v + 2*(lane/16)]
// ---------------------------------------------------------------------------
__global__ __launch_bounds__(128) void CA_qkv_gemm(
    const float* __restrict__ xs, const float* __restrict__ xt,
    const float* __restrict__ Wq, const float* __restrict__ Wk,
    const float* __restrict__ Wv,
    float* __restrict__ Qo, float* __restrict__ Ko, float* __restrict__ Vo)
{
    const int lane = threadIdx.x & 31;
    const int w    = blockIdx.x * 4 + (threadIdx.x >> 5);   // 0..3071
    const int mat  = w >> 10;                               // 0=Q 1=K 2=V
    const int rem  = w & 1023;
    const int rowT = rem >> 4;                              // 0..63 (covers B*L rows)
    const int colT = rem & 15;                              // 0..15

    const float* X = (mat == 0) ? xt : xs;
    const float* W = (mat == 0) ? Wq : (mat == 1 ? Wk : Wv);
    float*       Y = (mat == 0) ? Qo : (mat == 1 ? Ko : Vo);

    const int lo = lane & 15;
    const int hi = lane >> 4;
    const int row0 = rowT * 16;
    const int e0   = colT * 16;

    const float* xrow = X + (size_t)(row0 + lo) * D_MODEL + 2 * hi;
    const float* wrow = W + (size_t)(e0   + lo) * D_MODEL + 2 * hi;

    v8f c = {};
    #pragma unroll 4
    for (int d0 = 0; d0 < D_MODEL; d0 += 4) {
        v2f a = *(const v2f*)(xrow + d0);
        v2f b = *(const v2f*)(wrow + d0);
        c = __builtin_amdgcn_wmma_f32_16x16x4_f32(false, a, false, b,
                                                  (short)0, c, false, false);
    }
    #pragma unroll
    for (int r = 0; r < 8; ++r) {
        int m = r + 8 * hi;                                 // C layout row
        Y[(size_t)(row0 + m) * D_MODEL + e0 + lo] = c[r];
    }
}

// ---------------------------------------------------------------------------
// Kernel 2: fused scores + online softmax + att x V.
// One wave handles (batch b, 16-row i-tile, head). Accumulator in WMMA C layout.
// ---------------------------------------------------------------------------
__global__ __launch_bounds__(128) void CA_attn(
    const float* __restrict__ Q, const float* __restrict__ K,
    const float* __restrict__ V,
    const float* __restrict__ w_mlp, const float* __restrict__ b_mlp,
    float* __restrict__ out)
{
    __shared__ float lds[4 * 512];          // per wave: 256 Q-tile + 256 P-tile
    const int lane = threadIdx.x & 31;
    const int wib  = threadIdx.x >> 5;
    float* Qs = lds + wib * 512;
    float* Ps = Qs + 256;

    const int w    = blockIdx.x * 4 + wib;  // 0..1023
    const int b    = w >> 9;
    const int rem  = w & 511;
    const int iT   = rem >> 4;              // 0..31
    const int head = rem & 15;

    const int lo = lane & 15;
    const int hi = lane >> 4;
    const int i0 = iT * 16;

    // Q tile (16 rows x 16 group elems) -> wave-private LDS
    #pragma unroll
    for (int t = 0; t < 8; ++t) {
        int idx = lane + 32 * t;            // 0..255
        int il = idx >> 4, g = idx & 15;
        Qs[idx] = Q[(size_t)(b * L_SEQ + i0 + il) * D_MODEL + head * NGROUP + g];
    }
    asm volatile("s_wait_dscnt 0" ::: "memory"); // wave-private LDS: ds in-order, just drain

    float wreg[16];
    #pragma unroll
    for (int g = 0; g < 16; ++g) wreg[g] = w_mlp[g];
    const float bias = b_mlp[0];

    float mrow[8], lrow[8];
    #pragma unroll
    for (int r = 0; r < 8; ++r) { mrow[r] = -3.0e38f; lrow[r] = 0.0f; }
    v8f acc = {};

    for (int j0 = 0; j0 < L_SEQ; j0 += 16) {
        // K rows: lane lo owns source row j0+lo (both half-waves duplicate)
        float kreg[16];
        const float* kp = K + (size_t)(b * L_SEQ + j0 + lo) * D_MODEL + head * NGROUP;
        #pragma unroll
        for (int g = 0; g < 16; ++g) kreg[g] = kp[g];

        // scores in C layout: entry r -> (i = r+8*hi, j = lo)
        float s[8];
        #pragma unroll
        for (int r = 0; r < 8; ++r) s[r] = 0.0f;
        #pragma unroll
        for (int g = 0; g < 16; ++g) {
            float kg = kreg[g], wg = wreg[g];
            #pragma unroll
            for (int r = 0; r < 8; ++r) {
                float d = Qs[(r + 8 * hi) * 16 + g] - kg;
                s[r] += fmaxf(d, 0.0f) * wg;
            }
        }
        #pragma unroll
        for (int r = 0; r < 8; ++r) s[r] = fmaxf(s[r] + bias, 0.0f);

        // online softmax: row = 16 lanes of one half-wave (xor masks stay in-half)
        #pragma unroll
        for (int r = 0; r < 8; ++r) {
            float mx = s[r];
            mx = fmaxf(mx, __shfl_xor(mx, 1));
            mx = fmaxf(mx, __shfl_xor(mx, 2));
            mx = fmaxf(mx, __shfl_xor(mx, 4));
            mx = fmaxf(mx, __shfl_xor(mx, 8));
            float mnew  = fmaxf(mrow[r], mx);
            float alpha = __expf(mrow[r] - mnew);
            float p     = __expf(s[r] - mnew);
            float ps = p;
            ps += __shfl_xor(ps, 1);
            ps += __shfl_xor(ps, 2);
            ps += __shfl_xor(ps, 4);
            ps += __shfl_xor(ps, 8);
            lrow[r] = lrow[r] * alpha + ps;
            mrow[r] = mnew;
            acc[r] *= alpha;                        // rescale running output
            Ps[(r + 8 * hi) * 16 + lo] = p;         // stage P (C layout -> LDS)
        }
        asm volatile("s_wait_dscnt 0" ::: "memory");

        // acc(16i x 16g) += P(16i x 16j) x Vt(16j x 16g): 4 chained 16x16x4 WMMAs
        // A layout read: a[v] = P[i=lo][k = 4*st + v + 2*hi]
        // B layout read: b[v] = V[b, j0 + k, g=lo*16 + head]
        #pragma unroll
        for (int st = 0; st < 4; ++st) {
            int kk = 4 * st + 2 * hi;
            v2f a, bm;
            a.x = Ps[lo * 16 + kk];
            a.y = Ps[lo * 16 + kk + 1];
            bm.x = V[(size_t)(b * L_SEQ + j0 + kk)     * D_MODEL + lo * NGROUP + head];
            bm.y = V[(size_t)(b * L_SEQ + j0 + kk + 1) * D_MODEL + lo * NGROUP + head];
            acc = __builtin_amdgcn_wmma_f32_16x16x4_f32(false, a, false, bm,
                                                        (short)0, acc, false, false);
        }
    }

    // normalize and store: C layout row m = r+8*hi, col g = lo; d = g*16 + head
    #pragma unroll
    for (int r = 0; r < 8; ++r) {
        int m = r + 8 * hi;
        out[(size_t)(b * L_SEQ + i0 + m) * D_MODEL + lo * NGROUP + head] =
            acc[r] / lrow[r];
    }
}

// ---------------------------------------------------------------------------
extern "C" void kernel_launch(void* const* d_in, const int* in_sizes, int n_in,
                              void* d_out, int out_size, void* d_ws, size_t ws_size,
                              hipStream_t stream) {
    const float* xs = (const float*)d_in[0];   // x_source (B,L2,D)
    const float* xt = (const float*)d_in[1];   // x_target (B,L1,D)
    const float* Wq = (const float*)d_in[2];
    const float* Wk = (const float*)d_in[3];
    const float* Wv = (const float*)d_in[4];
    const float* wm = (const float*)d_in[5];   // w_mlp (16)
    const float* bm = (const float*)d_in[6];   // b_mlp (1)

    float* Qw = (float*)d_ws;                  // 3 MB scratch: Q,K,V f32
    float* Kw = Qw + (size_t)NB * L_SEQ * D_MODEL;
    float* Vw = Kw + (size_t)NB * L_SEQ * D_MODEL;

    // 3 matrices * 64 row-tiles * 16 col-tiles = 3072 waves, 4 waves/block
    CA_qkv_gemm<<<768, 128, 0, stream>>>(xs, xt, Wq, Wk, Wv, Qw, Kw, Vw);
    // 2 batch * 32 i-tiles * 16 heads = 1024 waves, 4 waves/block
    CA_attn<<<256, 128, 0, stream>>>(Qw, Kw, Vw, wm, bm, (float*)d_out);
}